// Canny_Encoder_35098472743006
// MI455X (gfx1250) — compile-verified
//
#include <hip/hip_runtime.h>
#include <hip/hip_bf16.h>
#include <stdint.h>

typedef __attribute__((ext_vector_type(2))) float v2f;
typedef __attribute__((ext_vector_type(8))) float v8f;

#define BATCH 64
#define Hd 512
#define Wd 512
#define NPIX (BATCH * Hd * Wd)

// Gaussian 7-tap, sigma=1.5, normalized (computed in double, cast to float)
#define GW0 0.036632843f
#define GW1 0.111280762f
#define GW2 0.216745317f
#define GW3 0.270682152f

__device__ __forceinline__ int reflect(int i, int n) {
    if (i < 0) i = -i;
    if (i >= n) i = 2 * n - 2 - i;
    return i;
}

// band value g[t] for t in [0,7), else 0 — pure VALU select chain, no memory
__device__ __forceinline__ float gband(int t) {
    float r = 0.0f;
    r = (t == 0 || t == 6) ? GW0 : r;
    r = (t == 1 || t == 5) ? GW1 : r;
    r = (t == 2 || t == 4) ? GW2 : r;
    r = (t == 3) ? GW3 : r;
    return r;
}

// order-preserving float <-> uint for atomic min/max
__device__ __forceinline__ unsigned f2o(float f) {
    unsigned u = __float_as_uint(f);
    return (u & 0x80000000u) ? ~u : (u | 0x80000000u);
}
__device__ __forceinline__ float o2f(unsigned m) {
    unsigned u = (m & 0x80000000u) ? (m & 0x7FFFFFFFu) : ~m;
    return __uint_as_float(u);
}

// ---------------- K0: init min/max cells ----------------
__global__ void k_init(unsigned* mm) {
    if (threadIdx.x == 0) { mm[0] = 0xFFFFFFFFu; mm[1] = 0u; }
}

// ---------------- K1: global min/max reduction ----------------
__global__ void __launch_bounds__(256) k_minmax(const float* __restrict__ x, unsigned* mm) {
    int tid = blockIdx.x * blockDim.x + threadIdx.x;
    int stride = gridDim.x * blockDim.x;
    float lo = 3.4e38f, hi = -3.4e38f;
    for (int i = tid; i < NPIX; i += stride) {
        float v = x[i];
        lo = fminf(lo, v);
        hi = fmaxf(hi, v);
    }
    for (int o = 16; o > 0; o >>= 1) {  // wave32 reduce
        lo = fminf(lo, __shfl_down(lo, o, 32));
        hi = fmaxf(hi, __shfl_down(hi, o, 32));
    }
    __shared__ float slo[8], shi[8];
    int w = threadIdx.x >> 5, l = threadIdx.x & 31;
    if (l == 0) { slo[w] = lo; shi[w] = hi; }
    __syncthreads();
    if (threadIdx.x == 0) {
        float flo = slo[0], fhi = shi[0];
        for (int i = 1; i < 8; i++) { flo = fminf(flo, slo[i]); fhi = fmaxf(fhi, shi[i]); }
        atomicMin(&mm[0], f2o(flo));
        atomicMax(&mm[1], f2o(fhi));
    }
}

// ---------------- K2: normalize + floor(clip) ----------------
__global__ void __launch_bounds__(256) k_norm(const float* __restrict__ x,
                                              const unsigned* __restrict__ mm,
                                              float* __restrict__ img) {
    int i = blockIdx.x * 256 + threadIdx.x;
    if (i >= NPIX) return;
    float mn = o2f(mm[0]), mx = o2f(mm[1]);
    float v = 255.0f * (x[i] - mn) / (mx - mn);
    v = fminf(fmaxf(v, 0.0f), 255.0f);
    img[i] = floorf(v);
}

// ---------------- K3: horizontal 7-tap Gaussian via WMMA f32 16x16x4 ----------------
// Per wave: strip of 8 tiles; each tile D(16x16) = A(16x24 rows, reflect) x B(24x16 band).
// B[c,j] = g[c-j] is lane-only: hoisted into 6 v2f register fragments per strip.
#define TPW 8
__global__ void __launch_bounds__(256) k_rowblur(const float* __restrict__ img,
                                                 float* __restrict__ out) {
    __shared__ float lA[8][16 * 25];  // per-wave 16x24 tile, stride 25 (conflict-free gathers)
    const int wv = threadIdx.x >> 5;
    const int lane = threadIdx.x & 31;
    const int half = lane >> 4;  // A: K-pair selector; C/D: M-half selector (ISA layout)
    const int mj = lane & 15;    // A: row M; B/D: column N

    int strip = blockIdx.x * 8 + wv;  // 8192 strips = 65536 tiles / 8
    int b = strip >> 7;               // 128 strips per image (32 tile-rows x 4 strips)
    int rem = strip & 127;
    int ti = rem >> 2;
    int tj0 = (rem & 3) * TPW;
    const float* src = img + (size_t)b * Hd * Wd;
    float* dst = out + (size_t)b * Hd * Wd;
    int row0 = ti * 16;

    // hoist B fragments (immediates -> cndmask chain, once per strip)
    v2f bf[6];
#pragma unroll
    for (int s = 0; s < 6; s++) {
        int k0 = 4 * s + 2 * half;
        bf[s].x = gband(k0 - mj);
        bf[s].y = gband(k0 + 1 - mj);
    }

    for (int u = 0; u < TPW; ++u) {
        int tj = tj0 + u;
        int col0 = tj * 16 - 3;
        // wave-private LDS fill (same-wave DS ops are in-order; no barrier needed)
        for (int e = lane; e < 16 * 24; e += 32) {
            int r = e / 24, c = e - 24 * r;
            int gc = reflect(col0 + c, Wd);
            lA[wv][r * 25 + c] = src[(row0 + r) * Wd + gc];
        }
        v8f acc = {0.f, 0.f, 0.f, 0.f, 0.f, 0.f, 0.f, 0.f};
#pragma unroll
        for (int s = 0; s < 6; s++) {
            int k0 = 4 * s + 2 * half;
            v2f a;
            a.x = lA[wv][mj * 25 + k0];
            a.y = lA[wv][mj * 25 + k0 + 1];
            acc = __builtin_amdgcn_wmma_f32_16x16x4_f32(false, a, false, bf[s],
                                                        (short)0, acc, false, false);
        }
        int col = tj * 16 + mj;
#pragma unroll
        for (int r = 0; r < 8; r++) {
            int row = row0 + r + 8 * half;  // C/D layout: VGPR r -> M=r / M=r+8
            dst[row * Wd + col] = acc[r];
        }
    }
}

// ---------------- K4: vertical 7-tap Gaussian + round-to-even ----------------
__global__ void __launch_bounds__(256) k_colblur(const float* __restrict__ tmp,
                                                 float* __restrict__ out) {
    int idx = blockIdx.x * 256 + threadIdx.x;
    if (idx >= NPIX) return;
    int j = idx & 511;
    int i = (idx >> 9) & 511;
    int b = idx >> 18;
    const float* src = tmp + (size_t)b * Hd * Wd;
    constexpr float g[7] = {GW0, GW1, GW2, GW3, GW2, GW1, GW0};
    float s = 0.0f;
#pragma unroll
    for (int t = 0; t < 7; t++) {
        int r = reflect(i + t - 3, Hd);
        s += g[t] * src[r * Wd + j];
    }
    out[idx] = rintf(s);  // jnp.round == round-half-even
}

// ---------------- K5: Sobel + |dx|+|dy| + NMS + double threshold ----------------
__global__ void __launch_bounds__(256) k_nms(const float* __restrict__ blur,
                                             unsigned char* __restrict__ lab,
                                             unsigned char* __restrict__ s0) {
    __shared__ float sb[36 * 37];  // 32x32 tile + halo2 (reflect), stride 37
    __shared__ float sm[34 * 35];  // mag on tile + halo1 (zero outside image)
    int tile = blockIdx.x;
    int b = tile >> 8;  // 16x16 tiles per image
    int rem = tile & 255;
    int ti = rem >> 4, tj = rem & 15;
    int i0 = ti * 32, j0 = tj * 32;
    const float* src = blur + (size_t)b * Hd * Wd;

    for (int e = threadIdx.x; e < 36 * 36; e += 256) {
        int r = e / 36, c = e - 36 * r;
        sb[r * 37 + c] = src[reflect(i0 - 2 + r, Hd) * Wd + reflect(j0 - 2 + c, Wd)];
    }
    __syncthreads();
    for (int e = threadIdx.x; e < 34 * 34; e += 256) {
        int r = e / 34, c = e - 34 * r;
        int gi = i0 - 1 + r, gj = j0 - 1 + c;
        float m = 0.0f;
        if (gi >= 0 && gi < Hd && gj >= 0 && gj < Wd) {
            const float* p = sb + (r + 1) * 37 + (c + 1);
            float dx = (p[-37 + 1] - p[-37 - 1]) + 2.f * (p[1] - p[-1]) + (p[37 + 1] - p[37 - 1]);
            float dy = (p[37 - 1] - p[-37 - 1]) + 2.f * (p[37] - p[-37]) + (p[37 + 1] - p[-37 + 1]);
            m = fabsf(dx) + fabsf(dy);
        }
        sm[r * 35 + c] = m;
    }
    __syncthreads();
    for (int e = threadIdx.x; e < 32 * 32; e += 256) {
        int r = e >> 5, c = e & 31;
        const float* p = sb + (r + 2) * 37 + (c + 2);
        float dx = (p[-37 + 1] - p[-37 - 1]) + 2.f * (p[1] - p[-1]) + (p[37 + 1] - p[37 - 1]);
        float dy = (p[37 - 1] - p[-37 - 1]) + 2.f * (p[37] - p[-37]) + (p[37 + 1] - p[-37 + 1]);
        float ax = fabsf(dx), ay = fabsf(dy);
        const float* mp = sm + (r + 1) * 35 + (c + 1);
        float mag = mp[0];
        bool horiz = ay < 0.41421356237309510f * ax;
        bool vert = !horiz && (ay > 2.41421356237309500f * ax);
        bool ss = dx * dy >= 0.0f;
        float n1, n2;
        if (horiz)      { n1 = mp[-1];      n2 = mp[1]; }
        else if (vert)  { n1 = mp[-35];     n2 = mp[35]; }
        else if (ss)    { n1 = mp[-35 - 1]; n2 = mp[35 + 1]; }
        else            { n1 = mp[-35 + 1]; n2 = mp[35 - 1]; }
        bool keep = (mag > n1) && (mag >= n2);
        bool strong = keep && (mag > 80.0f);
        bool weak = keep && (mag > 40.0f);
        size_t gidx = (size_t)b * Hd * Wd + (size_t)(i0 + r) * Wd + (j0 + c);
        lab[gidx] = (unsigned char)((weak ? 1 : 0) | (strong ? 2 : 0));
        s0[gidx] = strong ? 1 : 0;
    }
}

// ---------------- K6: hysteresis pass — tile-local closure in LDS, ping-pong global ----------------
__global__ void __launch_bounds__(256) k_hyst(const unsigned char* __restrict__ sin,
                                              const unsigned char* __restrict__ lab,
                                              unsigned char* __restrict__ sout) {
    __shared__ unsigned char st[66 * 66];
    __shared__ unsigned char wk[64 * 64];
    __shared__ int chg;
    int tile = blockIdx.x;
    int b = tile >> 6;  // 8x8 tiles per image
    int rem = tile & 63;
    int ti = rem >> 3, tj = rem & 7;
    int i0 = ti * 64, j0 = tj * 64;
    const unsigned char* sb = sin + (size_t)b * Hd * Wd;
    const unsigned char* lb = lab + (size_t)b * Hd * Wd;

    for (int e = threadIdx.x; e < 66 * 66; e += 256) {
        int r = e / 66, c = e - 66 * r;
        int gi = i0 - 1 + r, gj = j0 - 1 + c;
        unsigned char v = 0;  // zero pad outside image / halo snapshot from previous pass
        if (gi >= 0 && gi < Hd && gj >= 0 && gj < Wd) v = sb[gi * Wd + gj];
        st[e] = v;
    }
    for (int e = threadIdx.x; e < 64 * 64; e += 256) {
        int r = e >> 6, c = e & 63;
        wk[e] = lb[(i0 + r) * Wd + (j0 + c)] & 1;
    }
    __syncthreads();
    for (int it = 0; it < 256; ++it) {
        if (threadIdx.x == 0) chg = 0;
        __syncthreads();
        int any = 0;
        for (int e = threadIdx.x; e < 64 * 64; e += 256) {
            int r = e >> 6, c = e & 63;
            int p = (r + 1) * 66 + (c + 1);
            if (wk[e] && !st[p]) {
                unsigned nb = (unsigned)st[p - 67] | st[p - 66] | st[p - 65] |
                              st[p - 1] | st[p + 1] |
                              st[p + 65] | st[p + 66] | st[p + 67];
                if (nb) { st[p] = 1; any = 1; }  // monotone: unique fixed point
            }
        }
        if (any) atomicOr(&chg, 1);
        __syncthreads();
        if (!chg) break;  // uniform
        __syncthreads();  // separate read of chg from next-iter reset
    }
    unsigned char* so = sout + (size_t)b * Hd * Wd;
    for (int e = threadIdx.x; e < 64 * 64; e += 256) {
        int r = e >> 6, c = e & 63;
        so[(i0 + r) * Wd + (j0 + c)] = st[(r + 1) * 66 + (c + 1)];
    }
}

// ---------------- K7: plus-shaped dilation + cast to float ----------------
__global__ void __launch_bounds__(256) k_final(const unsigned char* __restrict__ s,
                                               float* __restrict__ out) {
    int idx = blockIdx.x * 256 + threadIdx.x;
    if (idx >= NPIX) return;
    int j = idx & 511, i = (idx >> 9) & 511, b = idx >> 18;
    const unsigned char* p = s + (size_t)b * Hd * Wd + (size_t)i * Wd + j;
    unsigned v = p[0];
    if (i > 0)      v |= p[-Wd];
    if (i < Hd - 1) v |= p[Wd];
    if (j > 0)      v |= p[-1];
    if (j < Wd - 1) v |= p[1];
    out[idx] = v ? 1.0f : 0.0f;
}

extern "C" void kernel_launch(void* const* d_in, const int* in_sizes, int n_in,
                              void* d_out, int out_size, void* d_ws, size_t ws_size,
                              hipStream_t stream) {
    const float* x = (const float*)d_in[0];
    float* out = (float*)d_out;

    char* ws = (char*)d_ws;
    size_t off = 0;
    unsigned* mm = (unsigned*)(ws + off);            off += 256;
    float* img = (float*)(ws + off);                 off += (size_t)NPIX * sizeof(float);
    unsigned char* lab = (unsigned char*)(ws + off); off += NPIX;
    unsigned char* sA = (unsigned char*)(ws + off);  off += NPIX;
    unsigned char* sB = (unsigned char*)(ws + off);  off += NPIX;

    const int nb = NPIX / 256;  // 65536

    k_init<<<1, 64, 0, stream>>>(mm);
    k_minmax<<<2048, 256, 0, stream>>>(x, mm);
    k_norm<<<nb, 256, 0, stream>>>(x, mm, img);
    // row blur: img -> d_out (scratch); 8192 wave-strips of 8 tiles, 8 waves per block
    k_rowblur<<<1024, 256, 0, stream>>>(img, out);
    // col blur + round: d_out -> img (img now holds "blurred")
    k_colblur<<<nb, 256, 0, stream>>>(out, img);
    // Sobel + NMS + thresholds: img -> lab (bit0 weak, bit1 strong), sA = strong seed
    k_nms<<<16384, 256, 0, stream>>>(img, lab, sA);
    // hysteresis: 32 ping-pong passes (even count -> final state in sA)
    for (int p = 0; p < 32; ++p) {
        const unsigned char* si = (p & 1) ? sB : sA;
        unsigned char* so = (p & 1) ? sA : sB;
        k_hyst<<<4096, 256, 0, stream>>>(si, lab, so);
    }
    k_final<<<nb, 256, 0, stream>>>(sA, out);
}